// ClusteringLayer_7215545057865
// MI455X (gfx1250) — compile-verified
//
#include <hip/hip_runtime.h>
#include <stdint.h>

// Problem constants (from reference): N=100000 points, K=256 clusters, F=256 features.
#define FDIM 256
#define KCLUST 256
#define TN 64            // points per workgroup
#define LDS_STRIDE 260   // 260 % 64 == 4 -> conflict-free b64 LDS reads of A fragments

typedef __attribute__((ext_vector_type(2))) float v2f;
typedef __attribute__((ext_vector_type(4))) float v4f;
typedef __attribute__((ext_vector_type(8))) float v8f;

// Map float to an unsigned with the same total order (for atomic u64 min).
__device__ __forceinline__ unsigned order_f32(float f) {
  unsigned b = __float_as_uint(f);
  return (b & 0x80000000u) ? ~b : (b | 0x80000000u);
}

__global__ void init_ws_kernel(unsigned long long* __restrict__ ws) {
  ws[threadIdx.x] = ~0ull;
}

__global__ void __launch_bounds__(256)
cluster_argmin_kernel(const float* __restrict__ x,
                      const float* __restrict__ centers,
                      unsigned long long* __restrict__ ws,
                      int npts) {
  __shared__ float lds_x[TN * LDS_STRIDE];
  __shared__ float lds_xsq[TN];
  __shared__ float lds_part[256];

  const int t = threadIdx.x;
  const int n0 = blockIdx.x * TN;

  // ---- Stage 64-point fp32 strip into LDS (coalesced b128 global loads),
  // ---- computing |x|^2 per point on the fly. OOB points get xsq=+inf.
  {
    const int p = t >> 2;          // point within strip
    const int q = t & 3;           // quarter of the feature row
    const int n = n0 + p;
    float sq = 0.0f;
    float* dst = &lds_x[p * LDS_STRIDE + q * 64];
    if (n < npts) {
      const v4f* src = (const v4f*)(x + (size_t)n * FDIM + q * 64);
#pragma unroll
      for (int j = 0; j < 16; ++j) {
        v4f v = src[j];
        *(v4f*)(dst + j * 4) = v;
        sq += v.x * v.x + v.y * v.y + v.z * v.z + v.w * v.w;
      }
    } else {
#pragma unroll
      for (int j = 0; j < 16; ++j) *(v4f*)(dst + j * 4) = (v4f)0.0f;
      sq = (q == 0) ? __builtin_huge_valf() : 0.0f;
    }
    lds_part[t] = sq;
  }
  __syncthreads();
  if (t < TN)
    lds_xsq[t] = lds_part[4 * t] + lds_part[4 * t + 1] +
                 lds_part[4 * t + 2] + lds_part[4 * t + 3];
  __syncthreads();

  // ---- fp32 WMMA GEMM: cross[p_tile(16) x c_tile(16)] accumulated over F.
  const int wave = t >> 5;
  const int lane = t & 31;
  const int m = lane & 15;       // row-within-tile for A, col/row for B (ISA layout)
  const int h = lane >> 4;       // K-half select
  const int kb = wave * 32;      // this wave owns clusters [kb, kb+32)

  v8f acc[4][2];
#pragma unroll
  for (int p = 0; p < 4; ++p) {
    acc[p][0] = (v8f)0.0f;
    acc[p][1] = (v8f)0.0f;
  }

  const float* c0 = centers + (size_t)(kb + m) * FDIM;        // B tile 0 row
  const float* c1 = centers + (size_t)(kb + 16 + m) * FDIM;   // B tile 1 row
  const float* lx = &lds_x[m * LDS_STRIDE];

#pragma unroll 4
  for (int kk = 0; kk < FDIM; kk += 4) {
    const int off = kk + 2 * h;
    // B fragments (4x16 f32, col n = lane%16, K = 2h+v): centers row, 2 floats
    v2f b0 = *(const v2f*)(c0 + off);
    v2f b1 = *(const v2f*)(c1 + off);
    // A fragments (16x4 f32, row m = lane%16, K = 2h+v): LDS x strip, 2 floats
    v2f a[4];
#pragma unroll
    for (int p = 0; p < 4; ++p)
      a[p] = *(const v2f*)(lx + p * 16 * LDS_STRIDE + off);
#pragma unroll
    for (int p = 0; p < 4; ++p) {
      acc[p][0] = __builtin_amdgcn_wmma_f32_16x16x4_f32(
          false, a[p], false, b0, (short)0, acc[p][0], false, false);
      acc[p][1] = __builtin_amdgcn_wmma_f32_16x16x4_f32(
          false, a[p], false, b1, (short)0, acc[p][1], false, false);
    }
  }

  // ---- Per-cluster argmin over this 64-point strip.
  // C layout: VGPR r, lane l -> M = r + 8*(l/16) (point), N = l%16 (cluster).
#pragma unroll
  for (int ct = 0; ct < 2; ++ct) {
    float best = __builtin_huge_valf();
    unsigned bidx = 0xFFFFFFFFu;
#pragma unroll
    for (int p = 0; p < 4; ++p) {
#pragma unroll
      for (int r = 0; r < 8; ++r) {
        const int mrow = p * 16 + r + 8 * h;
        const float s = lds_xsq[mrow] - 2.0f * acc[p][ct][r];
        const unsigned idx = (unsigned)(n0 + mrow);
        // idx increases monotonically within a lane -> '<' keeps first index on ties
        if (s < best) { best = s; bidx = idx; }
      }
    }
    unsigned long long key =
        ((unsigned long long)order_f32(best) << 32) | (unsigned long long)bidx;
    // merge the two M-halves (lane <-> lane^16 hold the other 8 rows, same cluster)
    {
      unsigned lo = (unsigned)key, hi = (unsigned)(key >> 32);
      unsigned olo = __shfl_xor(lo, 16, 32);
      unsigned ohi = __shfl_xor(hi, 16, 32);
      unsigned long long other = ((unsigned long long)ohi << 32) | (unsigned long long)olo;
      if (other < key) key = other;
    }
    if (h == 0) {
      // global_atomic_min_u64: min packed (score, index) across all blocks
      atomicMin(&ws[kb + ct * 16 + m], key);
    }
  }
}

__global__ void gather_kernel(const float* __restrict__ x,
                              const unsigned long long* __restrict__ ws,
                              float* __restrict__ out) {
  const int k = blockIdx.x;
  const unsigned idx = (unsigned)(ws[k] & 0xFFFFFFFFull);
  out[(size_t)k * FDIM + threadIdx.x] = x[(size_t)idx * FDIM + threadIdx.x];
}

extern "C" void kernel_launch(void* const* d_in, const int* in_sizes, int n_in,
                              void* d_out, int out_size, void* d_ws, size_t ws_size,
                              hipStream_t stream) {
  const float* x = (const float*)d_in[0];          // (1, N, 256) fp32
  const float* centers = (const float*)d_in[1];    // (256, 256) fp32
  float* out = (float*)d_out;                      // (1, 256, 256) fp32
  unsigned long long* ws = (unsigned long long*)d_ws;

  const int npts = in_sizes[0] / FDIM;             // N = 100000

  init_ws_kernel<<<1, KCLUST, 0, stream>>>(ws);
  const int nblocks = (npts + TN - 1) / TN;
  cluster_argmin_kernel<<<nblocks, 256, 0, stream>>>(x, centers, ws, npts);
  gather_kernel<<<KCLUST, FDIM, 0, stream>>>(x, ws, out);
}